// GNN_Conv_70325794505477
// MI455X (gfx1250) — compile-verified
//
#include <hip/hip_runtime.h>
#include <hip/hip_bf16.h>

#define GNN_N 50000
#define GNN_D 64
#define GNN_E 800000

typedef float v2f __attribute__((ext_vector_type(2)));
typedef float v4f __attribute__((ext_vector_type(4)));
typedef float v8f __attribute__((ext_vector_type(8)));

// -------------------------------------------------------------------------
// Degree count: one thread per edge. Runs once; both layers reuse cnt.
// -------------------------------------------------------------------------
__global__ __launch_bounds__(256)
void sage_degree(const int* __restrict__ dst, float* __restrict__ cnt) {
  int e = blockIdx.x * blockDim.x + threadIdx.x;
  if (e >= GNN_E) return;
  atomicAdd(&cnt[dst[e]], 1.0f);
}

// -------------------------------------------------------------------------
// Scatter: agg[dst[e]][4g..4g+3] += feat[src[e]][4g..4g+3].
// One thread per (edge, 4-feature chunk): a single b128 gather (L2-resident,
// node table is 12.8MB << 192MB L2) + 4 f32 atomics. 4x fewer VMEM issue
// slots and 4x fewer edge-index loads than scalar-per-feature — this phase
// is L2-latency/atomic bound, so issue count is what matters.
// -------------------------------------------------------------------------
__global__ __launch_bounds__(256)
void sage_scatter(const float* __restrict__ feat,
                  const int*   __restrict__ src,
                  const int*   __restrict__ dst,
                  float* __restrict__ agg) {
  long long idx = (long long)blockIdx.x * blockDim.x + threadIdx.x;
  if (idx >= (long long)GNN_E * (GNN_D / 4)) return;
  int e = (int)(idx >> 4);        // D/4 == 16 chunks per edge
  int g = (int)(idx & 15);
  int s = src[e];
  int d = dst[e];
  v4f m = *(const v4f*)(feat + (size_t)s * GNN_D + g * 4);
  float* a = agg + (size_t)d * GNN_D + g * 4;
  atomicAdd(a + 0, m.x);
  atomicAdd(a + 1, m.y);
  atomicAdd(a + 2, m.z);
  atomicAdd(a + 3, m.w);
}

// -------------------------------------------------------------------------
// Fused SAGE layer GEMM using V_WMMA_F32_16X16X4_F32 (full fp32 precision —
// the matmul is ~1.6 GFLOP total and nowhere near the bottleneck, so we keep
// exact fp32 semantics instead of dropping to bf16/fp8).
//
//   out[16 x 64] = (agg/deg)[16 x 64] @ Wl^T + x[16 x 64] @ Wr^T + b (+tanh)
//
// Block = 128 threads = 4 waves; block owns a 16-node row tile, each wave a
// 16-wide column tile. K=64 in 16 steps of K=4 per weight matrix -> 32 wmma
// per wave.
//
// wave32 fragment layouts (ISA 7.12.2, 32-bit):
//   A 16x4 : {half=lane>>4, m=lane&15}; vgpr v holds A[m][k0+2*half+v]
//   B 4x16 : {half, n=lane&15};         vgpr v holds B[k0+2*half+v][n],
//            B[k][o] = W[o][k]  (computes @ W^T)
//   C 16x16: vgpr g holds C[g + 8*half][n]
// -------------------------------------------------------------------------
template <bool TANH>
__global__ __launch_bounds__(128)
void sage_gemm(const float* __restrict__ aggsum,
               const float* __restrict__ cnt,
               const float* __restrict__ xin,
               const float* __restrict__ Wl,
               const float* __restrict__ bl,
               const float* __restrict__ Wr,
               float* __restrict__ out) {
  const int node0 = blockIdx.x * 16;
  const int wave  = threadIdx.x >> 5;     // 0..3 -> output column tile
  const int lane  = threadIdx.x & 31;
  const int half  = lane >> 4;            // 0/1
  const int lr    = lane & 15;            // m for A, n for B/C
  const int obase = wave * 16;
  const int kofs  = half * 2;

  const int arow = node0 + lr;
  const float deg  = cnt[arow];
  const float rcnt = 1.0f / fmaxf(deg, 1.0f);   // mean with deg>=1 guard

  const float* __restrict__ Arow  = aggsum + (size_t)arow * GNN_D;
  const float* __restrict__ Xrow  = xin    + (size_t)arow * GNN_D;
  const float* __restrict__ WlRow = Wl + (size_t)(obase + lr) * GNN_D;
  const float* __restrict__ WrRow = Wr + (size_t)(obase + lr) * GNN_D;

  v8f acc = {};

  // term 1: mean(agg) @ Wl^T  (normalization folded into the A fragment)
#pragma unroll
  for (int k0 = 0; k0 < GNN_D; k0 += 4) {
    v2f a, b;
    a.x = Arow[k0 + kofs + 0] * rcnt;
    a.y = Arow[k0 + kofs + 1] * rcnt;
    b.x = WlRow[k0 + kofs + 0];
    b.y = WlRow[k0 + kofs + 1];
    acc = __builtin_amdgcn_wmma_f32_16x16x4_f32(
        /*neg_a=*/false, a, /*neg_b=*/false, b,
        /*c_mod=*/(short)0, acc, /*reuse_a=*/false, /*reuse_b=*/false);
  }

  // term 2: x @ Wr^T
#pragma unroll
  for (int k0 = 0; k0 < GNN_D; k0 += 4) {
    v2f a, b;
    a.x = Xrow[k0 + kofs + 0];
    a.y = Xrow[k0 + kofs + 1];
    b.x = WrRow[k0 + kofs + 0];
    b.y = WrRow[k0 + kofs + 1];
    acc = __builtin_amdgcn_wmma_f32_16x16x4_f32(
        false, a, false, b, (short)0, acc, false, false);
  }

  const float bias = bl[obase + lr];      // bias indexed by output column n
#pragma unroll
  for (int g = 0; g < 8; ++g) {
    const int m = g + 8 * half;
    float v = acc[g] + bias;
    if (TANH) v = tanhf(v);
    out[(size_t)(node0 + m) * GNN_D + obase + lr] = v;
  }
}

// -------------------------------------------------------------------------
// Host-side orchestration (graph-capture safe: only async ops on `stream`).
// Workspace layout: agg[N*D] | cnt[N] | h[N*D]   (~25.8 MB)
// -------------------------------------------------------------------------
extern "C" void kernel_launch(void* const* d_in, const int* in_sizes, int n_in,
                              void* d_out, int out_size, void* d_ws, size_t ws_size,
                              hipStream_t stream) {
  const float* x   = (const float*)d_in[0];
  const int*   ei  = (const int*)  d_in[1];   // [2, E] int32
  const float* Wl1 = (const float*)d_in[2];
  const float* b1  = (const float*)d_in[3];
  const float* Wr1 = (const float*)d_in[4];
  const float* Wl2 = (const float*)d_in[5];
  const float* b2  = (const float*)d_in[6];
  const float* Wr2 = (const float*)d_in[7];
  float* out = (float*)d_out;

  float* agg = (float*)d_ws;
  float* cnt = agg + (size_t)GNN_N * GNN_D;
  float* h   = cnt + GNN_N;

  const int* src = ei;           // row 0
  const int* dst = ei + GNN_E;   // row 1

  const long long stot = (long long)GNN_E * (GNN_D / 4);
  const int sblocks = (int)((stot + 255) / 256);
  const int dblocks = (GNN_E + 255) / 256;
  const int gblocks = GNN_N / 16;   // 3125

  // ---- layer 1 ----
  hipMemsetAsync(agg, 0, ((size_t)GNN_N * GNN_D + GNN_N) * sizeof(float), stream);
  sage_degree<<<dblocks, 256, 0, stream>>>(dst, cnt);
  sage_scatter<<<sblocks, 256, 0, stream>>>(x, src, dst, agg);
  sage_gemm<true><<<gblocks, 128, 0, stream>>>(agg, cnt, x, Wl1, b1, Wr1, h);

  // ---- layer 2 (degree counts are identical; only reset the numerator) ----
  hipMemsetAsync(agg, 0, (size_t)GNN_N * GNN_D * sizeof(float), stream);
  sage_scatter<<<sblocks, 256, 0, stream>>>(h, src, dst, agg);
  sage_gemm<false><<<gblocks, 128, 0, stream>>>(agg, cnt, h, Wl2, b2, Wr2, out);
}